// Classifier_sep_model_36687610642724
// MI455X (gfx1250) — compile-verified
//
#include <hip/hip_runtime.h>
#include <hip/hip_bf16.h>

// Problem constants (match reference)
#define BB 128
#define SS 2048
#define HH 768
#define LL 26

typedef __attribute__((ext_vector_type(2))) float v2f;
typedef __attribute__((ext_vector_type(8))) float v8f;

// ---------------------------------------------------------------------------
// Kernel 1: sep_ids[b] = min(sum_s mask[b,s], S-1)   (one block per row)
// ---------------------------------------------------------------------------
__global__ void sep_ids_kernel(const int* __restrict__ mask,
                               int* __restrict__ sep_ids) {
    const int b = blockIdx.x;
    const int tid = threadIdx.x;              // 256 threads = 8 waves (wave32)
    const int* row = mask + (size_t)b * SS;

    int sum = 0;
    for (int i = tid; i < SS; i += 256) sum += row[i];

    // wave32 shuffle reduce
    #pragma unroll
    for (int off = 16; off > 0; off >>= 1)
        sum += __shfl_down(sum, off, 32);

    __shared__ int smem[8];
    if ((tid & 31) == 0) smem[tid >> 5] = sum;
    __syncthreads();

    if (tid == 0) {
        int total = 0;
        #pragma unroll
        for (int w = 0; w < 8; ++w) total += smem[w];
        sep_ids[b] = (total < SS) ? total : (SS - 1);
    }
}

// ---------------------------------------------------------------------------
// Kernel 2: fused gather + GEMM via V_WMMA_F32_16X16X4_F32
//   pred[m,n] = sum_k outs[m,k] * W[n,k] + bias[n]
//   outs[m,k] = (k < H) ? hidden[m, sep_ids[m], k] : cls[m, k-H]
// Grid: (8, 2) blocks of 32 threads (1 wave each). Each wave -> 16x16 C tile.
// ---------------------------------------------------------------------------
__global__ void sep_gemm_wmma_kernel(const float* __restrict__ hidden,
                                     const float* __restrict__ cls,
                                     const float* __restrict__ W,
                                     const float* __restrict__ bias,
                                     const int* __restrict__ sep_ids,
                                     float* __restrict__ out) {
    const int lane = threadIdx.x;        // 0..31
    const int hi   = lane >> 4;          // 0 or 1 (lane half)
    const int lm   = lane & 15;

    const int m = blockIdx.x * 16 + lm;              // batch row for A frag
    const int n = blockIdx.y * 16 + lm;              // logical output column
    const int nc = (n < LL) ? n : (LL - 1);          // clamp for safe W loads

    const int sep = sep_ids[m];
    const float* hidRow = hidden + ((size_t)m * SS + (size_t)sep) * HH;
    const float* clsRow = cls + (size_t)m * HH;
    const float* wRow   = W + (size_t)nc * (2 * HH); // W[n, :], B[k,n]=W[n,k]

    const int koff = 2 * hi;             // per-lane K offset within a k-step
    v8f acc = {};

    // K = 0 .. H-1 : gathered hidden row
    #pragma unroll 8
    for (int j = 0; j < HH / 4; ++j) {
        const int k = 4 * j + koff;
        v2f a = *(const v2f*)(hidRow + k);
        v2f bv = *(const v2f*)(wRow + k);
        acc = __builtin_amdgcn_wmma_f32_16x16x4_f32(
            /*neg_a=*/false, a, /*neg_b=*/false, bv,
            /*c_mod=*/(short)0, acc, /*reuse_a=*/false, /*reuse_b=*/false);
    }
    // K = H .. 2H-1 : cls row
    #pragma unroll 8
    for (int j = 0; j < HH / 4; ++j) {
        const int k = 4 * j + koff;
        v2f a = *(const v2f*)(clsRow + k);
        v2f bv = *(const v2f*)(wRow + HH + k);
        acc = __builtin_amdgcn_wmma_f32_16x16x4_f32(
            false, a, false, bv, (short)0, acc, false, false);
    }

    // C/D layout: VGPR v, lane l -> M = v + 8*(l>>4), N = l&15
    if (n < LL) {
        const float bb = bias[n];
        #pragma unroll
        for (int v = 0; v < 8; ++v) {
            const int row = blockIdx.x * 16 + v + 8 * hi;
            out[(size_t)row * LL + n] = acc[v] + bb;
        }
    }
}

// ---------------------------------------------------------------------------
extern "C" void kernel_launch(void* const* d_in, const int* in_sizes, int n_in,
                              void* d_out, int out_size, void* d_ws, size_t ws_size,
                              hipStream_t stream) {
    const float* hidden = (const float*)d_in[0];   // [B,S,H] f32
    const float* cls    = (const float*)d_in[1];   // [B,H]   f32
    const int*   mask   = (const int*)  d_in[2];   // [B,S]   i32
    const float* W      = (const float*)d_in[3];   // [L,2H]  f32
    const float* bias   = (const float*)d_in[4];   // [L]     f32
    float* out          = (float*)d_out;           // [B,L]   f32

    int* sep_ids = (int*)d_ws;                     // 128 ints of scratch

    sep_ids_kernel<<<BB, 256, 0, stream>>>(mask, sep_ids);

    dim3 grid(BB / 16, 2);                          // (8, 2) tiles
    sep_gemm_wmma_kernel<<<grid, 32, 0, stream>>>(hidden, cls, W, bias,
                                                  sep_ids, out);
}